// ESN_60395830116936
// MI455X (gfx1250) — compile-verified
//
#include <hip/hip_runtime.h>
#include <hip/hip_bf16.h>

// Problem constants (from reference): x:(B,C,T), W_in:(R,1), d:(R,)
// out:(B,C,T,R)  s_t = tanh(x[b,c,t]*w[r] + s_{t-1}*d[r])
constexpr int kB = 32;
constexpr int kC = 8;
constexpr int kT = 512;
constexpr int kR = 256;

// ---- fast tanh: prefer gfx1250 v_tanh_f32 transcendental ----
__device__ __forceinline__ float fast_tanh(float v) {
#if __has_builtin(__builtin_amdgcn_tanhf)
    return __builtin_amdgcn_tanhf(v);
#elif __has_builtin(__builtin_amdgcn_tanh_f32)
    return __builtin_amdgcn_tanh_f32(v);
#else
    // tanh(x) = 1 - 2/(exp(2x)+1), exp(2x) = exp2(x * 2*log2(e))
    float e = __builtin_amdgcn_exp2f(v * 2.8853900817779268f);
    return 1.0f - 2.0f * __builtin_amdgcn_rcpf(e + 1.0f);
#endif
}

__global__ __launch_bounds__(kR) void ESN_kernel(const float* __restrict__ x,
                                                 const float* __restrict__ w,
                                                 const float* __restrict__ dvec,
                                                 float* __restrict__ out) {
    const int r  = threadIdx.x;   // 0..255  (reservoir index, fastest output dim)
    const int bc = blockIdx.x;    // 0..255  (b*C + c)

    const float wr = w[r];
    const float dr = dvec[r];

    const float* __restrict__ xrow = x + (size_t)bc * kT;
    float* __restrict__ orow       = out + (size_t)bc * kT * kR + r;

    // Stage the shared x[b,c,:] row (2 KiB) into LDS once; all 8 waves
    // broadcast-read it every step.
    __shared__ float xs[kT];

#if __has_builtin(__builtin_amdgcn_global_load_async_to_lds_b32)
    {
        typedef __attribute__((address_space(1))) int gint_t;  // global int*
        typedef __attribute__((address_space(3))) int lint_t;  // LDS int*
        // 512 dwords with 256 lanes -> 2 async ops per lane (ASYNCcnt-tracked)
        __builtin_amdgcn_global_load_async_to_lds_b32(
            (gint_t*)(xrow + r),        (lint_t*)(xs + r),        0, 0);
        __builtin_amdgcn_global_load_async_to_lds_b32(
            (gint_t*)(xrow + r + 256),  (lint_t*)(xs + r + 256),  0, 0);
#if __has_builtin(__builtin_amdgcn_s_wait_asynccnt)
        __builtin_amdgcn_s_wait_asynccnt(0);
#else
        asm volatile("s_wait_asynccnt 0x0" ::: "memory");
#endif
    }
#else
    xs[r]       = xrow[r];
    xs[r + 256] = xrow[r + 256];
#endif
    __syncthreads();

    float s = 0.0f;
#pragma unroll 8
    for (int t = 0; t < kT; ++t) {
        const float u = xs[t];                       // LDS broadcast read
        s = fast_tanh(__builtin_fmaf(s, dr, u * wr));
        // streaming 128 MiB output: non-temporal, coalesced 128B/wave
        __builtin_nontemporal_store(s, orow + (size_t)t * kR);
    }
}

extern "C" void kernel_launch(void* const* d_in, const int* in_sizes, int n_in,
                              void* d_out, int out_size, void* d_ws, size_t ws_size,
                              hipStream_t stream) {
    const float* x    = (const float*)d_in[0];  // (B,C,T)
    const float* w    = (const float*)d_in[1];  // (R,1) -> flat (R,)
    const float* dvec = (const float*)d_in[2];  // (R,)
    float* out        = (float*)d_out;          // (B,C,T,R)

    dim3 grid(kB * kC);   // 256 blocks, one per (b,c)
    dim3 block(kR);       // 256 threads = 8 wave32s
    ESN_kernel<<<grid, block, 0, stream>>>(x, w, dvec, out);
}